// PiecewisePooling1D_38354057953797
// MI455X (gfx1250) — compile-verified
//
#include <hip/hip_runtime.h>

typedef float v4f __attribute__((ext_vector_type(4)));

constexpr int kB = 32;
constexpr int kL = 8192;
constexpr int kD = 128;
constexpr int kP = 8;
constexpr int kChunks = 4;            // grid = B*P*kChunks = 1024 workgroups
constexpr int kWaves  = 8;            // 256 threads = 8 wave32
constexpr int kBlock  = kWaves * 32;
constexpr int kD4     = kD / 4;       // 32 float4 per row (one row per wave)
constexpr int kDepth  = 4;            // async pipeline depth per wave
constexpr int kRowB   = kD * 4;       // 512 bytes per row

__device__ __forceinline__ v4f vmax4(v4f a, v4f b) {
  v4f r;
  r[0] = fmaxf(a[0], b[0]);
  r[1] = fmaxf(a[1], b[1]);
  r[2] = fmaxf(a[2], b[2]);
  r[3] = fmaxf(a[3], b[3]);
  return r;
}

// Sign-aware float atomic max (dest pre-initialized to -inf).
__device__ __forceinline__ void atomic_max_f32(float* addr, float v) {
  int iv = __float_as_int(v);
  if (v == -__builtin_inff()) return;
  if (iv >= 0) {
    atomicMax(reinterpret_cast<int*>(addr), iv);
  } else {
    atomicMin(reinterpret_cast<unsigned int*>(addr), __float_as_uint(v));
  }
}

__global__ __launch_bounds__(256) void pp_init(float* __restrict__ out, int n) {
  int i = blockIdx.x * 256 + threadIdx.x;
  if (i < n) out[i] = -__builtin_inff();
}

__global__ __launch_bounds__(kBlock) void pp_pool(const float* __restrict__ x,
                                                  const int* __restrict__ pos,
                                                  float* __restrict__ out) {
  const int wg = (int)blockIdx.x;      // [0, B*P*kChunks)
  const int c  = wg & (kChunks - 1);
  const int bp = wg >> 2;              // b*kP + p
  const int p  = bp & (kP - 1);
  const int b  = bp >> 3;

  // Segment [start, end) of (b,p); this block handles chunk c.
  const int end   = pos[bp];
  const int start = (p == 0) ? 0 : pos[bp - 1];
  const int clen  = (end - start + kChunks - 1) >> 2;
  const int t0    = start + c * clen;
  int t1 = t0 + clen;
  if (t1 > end) t1 = end;

  const int wave = (int)threadIdx.x >> 5;
  const int lane = (int)threadIdx.x & 31;

  // Per-wave circular staging buffers (async engine writes, wave ds_loads read).
  __shared__ v4f stage[kWaves][kDepth][32];   // 16 KB
  __shared__ v4f red[kWaves][32];             //  4 KB

  // Global byte address of this lane's 16B within row t: xbase + t*512.
  const char* xbase =
      reinterpret_cast<const char*>(x + (size_t)b * kL * kD) + (size_t)lane * 16;

  // Workgroup-relative LDS byte address of slot 0 for this (wave, lane):
  // low 32 bits of a flat pointer into LDS are the LDS offset.
  const unsigned int ldsBase =
      (unsigned int)(uintptr_t)&stage[wave][0][lane];

  v4f acc = { -__builtin_inff(), -__builtin_inff(),
              -__builtin_inff(), -__builtin_inff() };

  int t    = t0 + wave;    // next row to consume (wave-strided)
  int tIss = t;            // next row to issue
  unsigned long long gAddr =
      (unsigned long long)(uintptr_t)xbase + (unsigned long long)tIss * kRowB;

  // ---- Prologue: fill the 4-deep async pipeline (ASYNCcnt-tracked) ----
#pragma unroll
  for (int k = 0; k < kDepth; ++k) {
    if (tIss < t1) {
      unsigned int la = ldsBase + (unsigned int)k * (unsigned int)(32 * sizeof(v4f));
      asm volatile("global_load_async_to_lds_b128 %0, %1, off"
                   :: "v"(la), "v"(gAddr) : "memory");
      tIss += kWaves;
      gAddr += (unsigned long long)kWaves * kRowB;
    }
  }

  // ---- Steady state: wait oldest, consume, reissue into freed slot ----
  int buf = 0;
  while (tIss < t1) {
    int tp = tIss + 4 * kWaves;                 // L2 prefetch ahead of the stream
    if (tp > end - 1) tp = end - 1;
    __builtin_prefetch(xbase + (size_t)tp * kRowB, 0, 0);  // global_prefetch_b8

    asm volatile("s_wait_asynccnt 0x3" ::: "memory");      // oldest slot landed
    v4f v = stage[wave][buf][lane];                        // ds_load_b128
    asm volatile("s_wait_dscnt 0x0" ::: "memory");         // data in VGPRs before reuse
    unsigned int la = ldsBase + (unsigned int)buf * (unsigned int)(32 * sizeof(v4f));
    asm volatile("global_load_async_to_lds_b128 %0, %1, off"
                 :: "v"(la), "v"(gAddr) : "memory");
    acc = vmax4(acc, v);

    gAddr += (unsigned long long)kWaves * kRowB;
    tIss  += kWaves;
    t     += kWaves;
    buf    = (buf + 1) & (kDepth - 1);
  }

  // ---- Drain: everything outstanding is done after one wait ----
  asm volatile("s_wait_asynccnt 0x0" ::: "memory");
  while (t < t1) {
    acc = vmax4(acc, stage[wave][buf][lane]);
    buf = (buf + 1) & (kDepth - 1);
    t  += kWaves;
  }

  // ---- Cross-wave reduction: 8 partial rows of 128 floats in LDS ----
  red[wave][lane] = acc;
  __syncthreads();

  if (threadIdx.x < kD) {
    const float* rf = reinterpret_cast<const float*>(red);
    float m = rf[threadIdx.x];
#pragma unroll
    for (int w = 1; w < kWaves; ++w) m = fmaxf(m, rf[w * kD + threadIdx.x]);
    atomic_max_f32(out + (size_t)bp * kD + threadIdx.x, m);
  }
}

extern "C" void kernel_launch(void* const* d_in, const int* in_sizes, int n_in,
                              void* d_out, int out_size, void* d_ws, size_t ws_size,
                              hipStream_t stream) {
  const float* x   = (const float*)d_in[0];   // [B, L, D] fp32
  const int*   pos = (const int*)d_in[1];     // [B, P] int32
  float*       out = (float*)d_out;           // [B, P, D] fp32

  const int nOut = kB * kP * kD;              // 32768
  pp_init<<<(nOut + 255) / 256, 256, 0, stream>>>(out, nOut);
  pp_pool<<<kB * kP * kChunks, kBlock, 0, stream>>>(x, pos, out);
}